// GATAE_40621800685687
// MI455X (gfx1250) — compile-verified
//
#include <hip/hip_runtime.h>
#include <hip/hip_bf16.h>

// ---------------------------------------------------------------------------
// GAT autoencoder for MI455X (gfx1250).
// Dense feature transforms use V_WMMA_F32_16X16X4_F32 (exact f32 math, K=4/2).
// Edge softmax/scatter passes are the memory-bound hot path: coalesced float4
// gathers of h[src] and GLOBAL_ATOMIC_ADD_F32 scatters, one edge per 32 lanes.
// All indexing is 32-bit (max index N*128 = 6.4M) to avoid 64-bit VALU noise.
// ---------------------------------------------------------------------------

typedef __attribute__((ext_vector_type(2))) float v2f;
typedef __attribute__((ext_vector_type(8))) float v8f;

#define EPS_BN 1e-5f
#define NEG_SLOPE 0.2f

// ---- monotonic float<->uint encoding for atomic float max --------------------
__device__ __forceinline__ unsigned fenc(float f) {
    unsigned u = __float_as_uint(f);
    return (u & 0x80000000u) ? ~u : (u | 0x80000000u);
}
__device__ __forceinline__ float fdec(unsigned k) {
    return (k & 0x80000000u) ? __uint_as_float(k & 0x7fffffffu)
                             : __uint_as_float(~k);
}

// ---- fills -------------------------------------------------------------------
__global__ void fill_f32(float* __restrict__ p, float v, int n) {
    int i = blockIdx.x * blockDim.x + threadIdx.x;
    if (i < n) p[i] = v;
}
__global__ void fill_u32(unsigned* __restrict__ p, unsigned v, int n) {
    int i = blockIdx.x * blockDim.x + threadIdx.x;
    if (i < n) p[i] = v;
}

// ---- BatchNorm: stats (sum, sumsq per channel) -------------------------------
__global__ __launch_bounds__(256) void bn_stats(const float* __restrict__ x,
                                                float* __restrict__ stats, int N) {
    __shared__ float lds[256][8];
    float s[4] = {0, 0, 0, 0}, q[4] = {0, 0, 0, 0};
    for (int i = blockIdx.x * blockDim.x + threadIdx.x; i < N;
         i += gridDim.x * blockDim.x) {
        float4 v = ((const float4*)x)[i];
        s[0] += v.x; s[1] += v.y; s[2] += v.z; s[3] += v.w;
        q[0] += v.x * v.x; q[1] += v.y * v.y; q[2] += v.z * v.z; q[3] += v.w * v.w;
    }
    for (int j = 0; j < 4; ++j) { lds[threadIdx.x][j] = s[j]; lds[threadIdx.x][4 + j] = q[j]; }
    __syncthreads();
    for (int off = 128; off > 0; off >>= 1) {
        if ((int)threadIdx.x < off)
            for (int j = 0; j < 8; ++j) lds[threadIdx.x][j] += lds[threadIdx.x + off][j];
        __syncthreads();
    }
    if (threadIdx.x < 8) atomicAdd(&stats[threadIdx.x], lds[0][threadIdx.x]);
}

__global__ void bn_apply(const float* __restrict__ x, const float* __restrict__ stats,
                         const float* __restrict__ gamma, const float* __restrict__ beta,
                         float* __restrict__ xin, int N) {
    int i = blockIdx.x * blockDim.x + threadIdx.x;
    if (i >= N * 4) return;
    int c = i & 3;
    float mu  = stats[c] / (float)N;
    float var = stats[4 + c] / (float)N - mu * mu;
    xin[i] = (x[i] - mu) * rsqrtf(var + EPS_BN) * gamma[c] + beta[c];
}

// ---- WMMA feature transform: Hout[N,128] = X[N,Kd] @ W[Kd,128], Kd in {2,4} --
// One block = 16 nodes x 128 cols; 8 waves, each wave one 16x16x4 f32 WMMA.
// Kd is a template param so zero-padding of the K dimension folds at compile
// time; stores use a single uniform tail check so the fast path is 8 plain
// global_store_b32 with immediate offsets (row stride 512 B).
template <int Kd>
__global__ __launch_bounds__(256) void feat4_wmma(const float* __restrict__ X,
                                                  const float* __restrict__ W,
                                                  float* __restrict__ Hout, int N) {
    const int wave = threadIdx.x >> 5;
    const int lane = threadIdx.x & 31;
    const int row  = lane & 15;   // M (for A) / N-col (for B,C,D)
    const int hi   = lane >> 4;   // 0 -> K={0,1}, 1 -> K={2,3}
    const int node0 = blockIdx.x * 16;
    const int col   = wave * 16 + row;
    const int kb    = hi * 2;

    int nodeA = node0 + row;
    if (nodeA >= N) nodeA = N - 1;   // clamp: keeps EXEC all-ones for WMMA

    v2f a, b;
    a.x = (kb + 0 < Kd) ? X[nodeA * Kd + kb + 0] : 0.0f;
    a.y = (kb + 1 < Kd) ? X[nodeA * Kd + kb + 1] : 0.0f;
    b.x = (kb + 0 < Kd) ? W[(kb + 0) * 128 + col] : 0.0f;
    b.y = (kb + 1 < Kd) ? W[(kb + 1) * 128 + col] : 0.0f;

    v8f c = {};
    c = __builtin_amdgcn_wmma_f32_16x16x4_f32(false, a, false, b, (short)0, c,
                                              false, false);

    float* base = Hout + (node0 + hi * 8) * 128 + col;  // rows node0+hi*8 .. +7
    if (node0 + 16 <= N) {
#pragma unroll
        for (int r = 0; r < 8; ++r) base[r * 128] = c[r];
    } else {
#pragma unroll
        for (int r = 0; r < 8; ++r)
            if (node0 + hi * 8 + r < N) base[r * 128] = c[r];
    }
}

// ---- scalar matvec: Y[N,Cout] = X[N,128] @ W[128,Cout] (Cout = 2 or 4) -------
__global__ void matvec128(const float* __restrict__ X, const float* __restrict__ W,
                          float* __restrict__ Y, int N, int Cout) {
    int t = blockIdx.x * blockDim.x + threadIdx.x;
    if (t >= N * Cout) return;
    int n = t / Cout, co = t % Cout;
    const float* xp = X + n * 128;
    float acc = 0.0f;
#pragma unroll 8
    for (int k = 0; k < 128; ++k) acc += xp[k] * W[k * Cout + co];
    Y[t] = acc;
}

// ---- alpha_s / alpha_d = sum_c h[n,h,c] * a_{src,dst}[h,c] -------------------
__global__ void alpha_kernel(const float* __restrict__ h, const float* __restrict__ a_s,
                             const float* __restrict__ a_d, float* __restrict__ as_o,
                             float* __restrict__ ad_o, int N, int H, int C) {
    int t = blockIdx.x * blockDim.x + threadIdx.x;
    if (t >= N * H) return;
    int n = t / H, hh = t % H;
    const float* hp  = h + (n * H + hh) * C;
    const float* asp = a_s + hh * C;
    const float* adp = a_d + hh * C;
    float ss = 0.0f, dd = 0.0f;
    for (int c = 0; c < C; ++c) { float v = hp[c]; ss += v * asp[c]; dd += v * adp[c]; }
    as_o[t] = ss; ad_o[t] = dd;
}

// ---- edge pass 1: segment max of leaky_relu(alpha_s[src]+alpha_d[dst]) -------
__global__ void edge_max_k(const float* __restrict__ aS, const float* __restrict__ aD,
                           const int* __restrict__ ei, int E, int N, int H,
                           unsigned* __restrict__ m) {
    int t = blockIdx.x * blockDim.x + threadIdx.x;
    int ET = E + N;
    if (t >= ET * H) return;
    int e = t / H, h = t % H;
    int s = (e < E) ? ei[e]     : (e - E);
    int d = (e < E) ? ei[E + e] : (e - E);
    float v = aS[s * H + h] + aD[d * H + h];
    v = (v >= 0.0f) ? v : NEG_SLOPE * v;
    atomicMax(&m[d * H + h], fenc(v));
}

// ---- edge pass 2: segment sum of exp(e - m[dst]) -----------------------------
__global__ void edge_sum_k(const float* __restrict__ aS, const float* __restrict__ aD,
                           const unsigned* __restrict__ m, const int* __restrict__ ei,
                           int E, int N, int H, float* __restrict__ ssum) {
    int t = blockIdx.x * blockDim.x + threadIdx.x;
    int ET = E + N;
    if (t >= ET * H) return;
    int e = t / H, h = t % H;
    int s = (e < E) ? ei[e]     : (e - E);
    int d = (e < E) ? ei[E + e] : (e - E);
    float v = aS[s * H + h] + aD[d * H + h];
    v = (v >= 0.0f) ? v : NEG_SLOPE * v;
    float ex = __expf(v - fdec(m[d * H + h]));
    atomicAdd(&ssum[d * H + h], ex);
}

// ---- edge pass 3 (wide, H=4, C=32): one edge per 32 lanes, float4 per lane ---
__global__ void edge_scatter128(const float* __restrict__ aS, const float* __restrict__ aD,
                                const unsigned* __restrict__ m, const float* __restrict__ ssum,
                                const float* __restrict__ hbuf, const int* __restrict__ ei,
                                int E, int N, float* __restrict__ out) {
    int gid = blockIdx.x * blockDim.x + threadIdx.x;
    int lane = gid & 31;
    int e = gid >> 5;
    if (e >= E + N) return;
    int s = (e < E) ? ei[e]     : (e - E);
    int d = (e < E) ? ei[E + e] : (e - E);
    int h = lane >> 3;  // 4 channels/lane, 32 channels/head -> head = lane/8
    float v = aS[s * 4 + h] + aD[d * 4 + h];
    v = (v >= 0.0f) ? v : NEG_SLOPE * v;
    float ex = __expf(v - fdec(m[d * 4 + h]));
    float a = ex / (ssum[d * 4 + h] + 1e-16f);
    const float4 hv = *(const float4*)(hbuf + s * 128 + lane * 4);
    float* op = out + d * 128 + lane * 4;
    atomicAdd(op + 0, hv.x * a);
    atomicAdd(op + 1, hv.y * a);
    atomicAdd(op + 2, hv.z * a);
    atomicAdd(op + 3, hv.w * a);
}

// ---- edge pass 3 (small, H=1, C<=4): one edge per thread ---------------------
__global__ void edge_scatter_small(const float* __restrict__ aS, const float* __restrict__ aD,
                                   const unsigned* __restrict__ m, const float* __restrict__ ssum,
                                   const float* __restrict__ hbuf, const int* __restrict__ ei,
                                   int E, int N, int C, float* __restrict__ out) {
    int e = blockIdx.x * blockDim.x + threadIdx.x;
    if (e >= E + N) return;
    int s = (e < E) ? ei[e]     : (e - E);
    int d = (e < E) ? ei[E + e] : (e - E);
    float v = aS[s] + aD[d];
    v = (v >= 0.0f) ? v : NEG_SLOPE * v;
    float ex = __expf(v - fdec(m[d]));
    float a = ex / (ssum[d] + 1e-16f);
    for (int c = 0; c < C; ++c)
        atomicAdd(&out[d * C + c], hbuf[s * C + c] * a);
}

// ---- bias (+optional ReLU) epilogue ------------------------------------------
__global__ void bias_act(float* __restrict__ y, const float* __restrict__ b,
                         int n, int dim, int relu) {
    int i = blockIdx.x * blockDim.x + threadIdx.x;
    if (i >= n) return;
    float v = y[i] + b[i % dim];
    if (relu) v = fmaxf(v, 0.0f);
    y[i] = v;
}

// ---------------------------------------------------------------------------
static inline int cdiv(long a, long b) { return (int)((a + b - 1) / b); }

extern "C" void kernel_launch(void* const* d_in, const int* in_sizes, int n_in,
                              void* d_out, int out_size, void* d_ws, size_t ws_size,
                              hipStream_t stream) {
    const float* x      = (const float*)d_in[0];
    const int*   ei     = (const int*)d_in[1];
    const float* bn_g   = (const float*)d_in[2];
    const float* bn_b   = (const float*)d_in[3];
    const float* w_e1   = (const float*)d_in[4];
    const float* as_e1  = (const float*)d_in[5];
    const float* ad_e1  = (const float*)d_in[6];
    const float* b_e1   = (const float*)d_in[7];
    const float* w_e3   = (const float*)d_in[8];
    const float* as_e3  = (const float*)d_in[9];
    const float* ad_e3  = (const float*)d_in[10];
    const float* b_e3   = (const float*)d_in[11];
    const float* w_d1   = (const float*)d_in[12];
    const float* as_d1  = (const float*)d_in[13];
    const float* ad_d1  = (const float*)d_in[14];
    const float* b_d1   = (const float*)d_in[15];
    const float* w_d3   = (const float*)d_in[16];
    const float* as_d3  = (const float*)d_in[17];
    const float* ad_d3  = (const float*)d_in[18];
    const float* b_d3   = (const float*)d_in[19];
    float* out = (float*)d_out;

    const int N  = in_sizes[0] / 4;
    const int E  = in_sizes[1] / 2;
    const int ET = E + N;

    // workspace layout (floats)
    float*    ws    = (float*)d_ws;
    float*    bufA  = ws;                              // [N,128] h of wide layers
    float*    bufB  = bufA + (long)N * 128;            // [N,128] out of wide layers
    float*    bufC  = bufB + (long)N * 128;            // [N,4]  x_in / small h
    float*    bufD  = bufC + (long)N * 4;              // [N,2]  latent
    float*    aS    = bufD + (long)N * 2;              // [N,4]
    float*    aD    = aS   + (long)N * 4;              // [N,4]
    unsigned* mBuf  = (unsigned*)(aD + (long)N * 4);   // [N,4]
    float*    sBuf  = (float*)mBuf + (long)N * 4;      // [N,4]
    float*    stats = sBuf + (long)N * 4;              // [8]

    const int T = 256;
    // encoding of -inf under fenc: bits(-inf)=0xFF800000, sign set -> ~u
    const unsigned M_INIT = 0x007FFFFFu;

    // ---------------- BatchNorm ----------------
    fill_f32<<<1, 8, 0, stream>>>(stats, 0.0f, 8);
    bn_stats<<<64, T, 0, stream>>>(x, stats, N);
    bn_apply<<<cdiv(N * 4, T), T, 0, stream>>>(x, stats, bn_g, bn_b, bufC, N);

    // ---------------- conv1: 4 -> 32x4heads, concat, ReLU ----------------
    feat4_wmma<4><<<cdiv(N, 16), T, 0, stream>>>(bufC, w_e1, bufA, N);
    alpha_kernel<<<cdiv(N * 4, T), T, 0, stream>>>(bufA, as_e1, ad_e1, aS, aD, N, 4, 32);
    fill_u32<<<cdiv(N * 4, T), T, 0, stream>>>(mBuf, M_INIT, N * 4);
    fill_f32<<<cdiv(N * 4, T), T, 0, stream>>>(sBuf, 0.0f, N * 4);
    fill_f32<<<cdiv((long)N * 128, T), T, 0, stream>>>(bufB, 0.0f, N * 128);
    edge_max_k<<<cdiv((long)ET * 4, T), T, 0, stream>>>(aS, aD, ei, E, N, 4, mBuf);
    edge_sum_k<<<cdiv((long)ET * 4, T), T, 0, stream>>>(aS, aD, mBuf, ei, E, N, 4, sBuf);
    edge_scatter128<<<cdiv((long)ET * 32, T), T, 0, stream>>>(aS, aD, mBuf, sBuf, bufA, ei, E, N, bufB);
    bias_act<<<cdiv((long)N * 128, T), T, 0, stream>>>(bufB, b_e1, N * 128, 128, 1);

    // ---------------- conv2: 128 -> 2, 1 head ----------------
    matvec128<<<cdiv(N * 2, T), T, 0, stream>>>(bufB, w_e3, bufC, N, 2);
    alpha_kernel<<<cdiv(N, T), T, 0, stream>>>(bufC, as_e3, ad_e3, aS, aD, N, 1, 2);
    fill_u32<<<cdiv(N, T), T, 0, stream>>>(mBuf, M_INIT, N);
    fill_f32<<<cdiv(N, T), T, 0, stream>>>(sBuf, 0.0f, N);
    fill_f32<<<cdiv(N * 2, T), T, 0, stream>>>(bufD, 0.0f, N * 2);
    edge_max_k<<<cdiv(ET, T), T, 0, stream>>>(aS, aD, ei, E, N, 1, mBuf);
    edge_sum_k<<<cdiv(ET, T), T, 0, stream>>>(aS, aD, mBuf, ei, E, N, 1, sBuf);
    edge_scatter_small<<<cdiv(ET, T), T, 0, stream>>>(aS, aD, mBuf, sBuf, bufC, ei, E, N, 2, bufD);
    bias_act<<<cdiv(N * 2, T), T, 0, stream>>>(bufD, b_e3, N * 2, 2, 0);

    // ---------------- conv3: 2 -> 32x4heads, concat, ReLU ----------------
    feat4_wmma<2><<<cdiv(N, 16), T, 0, stream>>>(bufD, w_d1, bufA, N);
    alpha_kernel<<<cdiv(N * 4, T), T, 0, stream>>>(bufA, as_d1, ad_d1, aS, aD, N, 4, 32);
    fill_u32<<<cdiv(N * 4, T), T, 0, stream>>>(mBuf, M_INIT, N * 4);
    fill_f32<<<cdiv(N * 4, T), T, 0, stream>>>(sBuf, 0.0f, N * 4);
    fill_f32<<<cdiv((long)N * 128, T), T, 0, stream>>>(bufB, 0.0f, N * 128);
    edge_max_k<<<cdiv((long)ET * 4, T), T, 0, stream>>>(aS, aD, ei, E, N, 4, mBuf);
    edge_sum_k<<<cdiv((long)ET * 4, T), T, 0, stream>>>(aS, aD, mBuf, ei, E, N, 4, sBuf);
    edge_scatter128<<<cdiv((long)ET * 32, T), T, 0, stream>>>(aS, aD, mBuf, sBuf, bufA, ei, E, N, bufB);
    bias_act<<<cdiv((long)N * 128, T), T, 0, stream>>>(bufB, b_d1, N * 128, 128, 1);

    // ---------------- conv4: 128 -> 4, 1 head ----------------
    matvec128<<<cdiv(N * 4, T), T, 0, stream>>>(bufB, w_d3, bufC, N, 4);
    alpha_kernel<<<cdiv(N, T), T, 0, stream>>>(bufC, as_d3, ad_d3, aS, aD, N, 1, 4);
    fill_u32<<<cdiv(N, T), T, 0, stream>>>(mBuf, M_INIT, N);
    fill_f32<<<cdiv(N, T), T, 0, stream>>>(sBuf, 0.0f, N);
    fill_f32<<<cdiv(N * 4, T), T, 0, stream>>>(out, 0.0f, N * 4);
    edge_max_k<<<cdiv(ET, T), T, 0, stream>>>(aS, aD, ei, E, N, 1, mBuf);
    edge_sum_k<<<cdiv(ET, T), T, 0, stream>>>(aS, aD, mBuf, ei, E, N, 1, sBuf);
    edge_scatter_small<<<cdiv(ET, T), T, 0, stream>>>(aS, aD, mBuf, sBuf, bufC, ei, E, N, 4, out);
    bias_act<<<cdiv(N * 4, T), T, 0, stream>>>(out, b_d3, N * 4, 4, 0);

    (void)n_in; (void)out_size; (void)ws_size;
}